// SolveSchedulingQP_86260123174050
// MI455X (gfx1250) — compile-verified
//
#include <hip/hip_runtime.h>

#define QP_N     192
#define QP_M1    191
#define QP_M     382
#define QP_NIT   20
#define QP_SIGMA 0.1f
#define QP_H     0.2f
#define BLK      256
#define NWAVES   (BLK / 32)

// ---- wave32 intra-wave reductions (no barriers, deterministic) ----
__device__ __forceinline__ float wave_sum(float v) {
  #pragma unroll
  for (int off = 16; off > 0; off >>= 1) v += __shfl_xor(v, off, 32);
  return v;
}
__device__ __forceinline__ float wave_min(float v) {
  #pragma unroll
  for (int off = 16; off > 0; off >>= 1) v = fminf(v, __shfl_xor(v, off, 32));
  return v;
}

// Read a PCR row with identity padding for out-of-range neighbors.
__device__ __forceinline__ void pcr_read(const float* A, const float* Bv,
                                         const float* Cv, const float* Dv,
                                         int idx, float& a, float& b, float& c, float& d) {
  if (idx >= 0 && idx < QP_N) { a = A[idx]; b = Bv[idx]; c = Cv[idx]; d = Dv[idx]; }
  else                        { a = 0.f;    b = 1.f;     c = 0.f;     d = 0.f;    }
}

__global__ __launch_bounds__(BLK)
void qp_ipm_kernel(const float* __restrict__ z0g, const float* __restrict__ mug,
                   const float* __restrict__ dgg, const float* __restrict__ d2gg,
                   float* __restrict__ out) {
  __shared__ float S0[QP_N], S1[QP_N], S2[QP_N], S3[QP_N];       // async input staging
  __shared__ float shz[QP_N], shv1[QP_N], shv2[QP_N], shw[QP_N], shdz[QP_N];
  __shared__ float wred[NWAVES];
  __shared__ float PA[2][QP_N], PB[2][QP_N], PC[2][QP_N], PD[2][QP_N];

  const int tid  = threadIdx.x;
  const int wid  = tid >> 5;
  const int lane = tid & 31;
  const long base = (long)blockIdx.x * QP_N;

  // ---- gfx1250 async global->LDS staging of this problem's 4 input rows ----
  if (tid < QP_N) {
    unsigned long long g0 = (unsigned long long)(const void*)(z0g  + base + tid);
    unsigned long long g1 = (unsigned long long)(const void*)(mug  + base + tid);
    unsigned long long g2 = (unsigned long long)(const void*)(dgg  + base + tid);
    unsigned long long g3 = (unsigned long long)(const void*)(d2gg + base + tid);
    // low 32 bits of a generic pointer to LDS == wave-relative LDS byte offset
    unsigned int l0 = (unsigned int)(unsigned long long)(const void*)&S0[tid];
    unsigned int l1 = (unsigned int)(unsigned long long)(const void*)&S1[tid];
    unsigned int l2 = (unsigned int)(unsigned long long)(const void*)&S2[tid];
    unsigned int l3 = (unsigned int)(unsigned long long)(const void*)&S3[tid];
    asm volatile("global_load_async_to_lds_b32 %0, %1, off" :: "v"(l0), "v"(g0) : "memory");
    asm volatile("global_load_async_to_lds_b32 %0, %1, off" :: "v"(l1), "v"(g1) : "memory");
    asm volatile("global_load_async_to_lds_b32 %0, %1, off" :: "v"(l2), "v"(g2) : "memory");
    asm volatile("global_load_async_to_lds_b32 %0, %1, off" :: "v"(l3), "v"(g3) : "memory");
    asm volatile("s_wait_asynccnt 0" ::: "memory");
  }
  __syncthreads();

  // ---- per-thread problem data ----
  float q = 1.f, p = 0.f, z = 0.f;
  if (tid < QP_N) {
    float vz0 = S0[tid], vmu = S1[tid], vdg = S2[tid], vd2 = S3[tid];
    q = vd2 + 1.0f;
    p = vdg - vd2 * vz0 - vmu;
    z = -p / q;                       // unconstrained warm start
    shz[tid] = z;
  }
  __syncthreads();

  float s_lo = 1.f, s_hi = 1.f, lam_lo = 1.f, lam_hi = 1.f;
  if (tid < QP_M1) {
    float d = shz[tid] - shz[tid + 1];          // (Dz)_i
    s_lo = fmaxf(QP_H - d, 1.0f);               // s = max(h - Gz, 1)
    s_hi = fmaxf(QP_H + d, 1.0f);
  }
  __syncthreads();

  for (int it = 0; it < QP_NIT; ++it) {
    if (tid < QP_N)  shz[tid]  = z;
    if (tid < QP_M1) shv1[tid] = lam_lo - lam_hi;   // for G^T lam = D^T(w1-w2)
    __syncthreads();

    float d = 0.f, rp_lo = 0.f, rp_hi = 0.f, sum = 0.f;
    if (tid < QP_M1) {
      d     = shz[tid] - shz[tid + 1];
      rp_lo = d  + s_lo - QP_H;
      rp_hi = -d + s_hi - QP_H;
      sum   = s_lo * lam_lo + s_hi * lam_hi;
    }
    float rd = 0.f;
    if (tid < QP_N) {
      float gtl = ((tid < QP_M1) ? shv1[tid] : 0.f) - ((tid > 0) ? shv1[tid - 1] : 0.f);
      rd = q * z + p + gtl;
    }
    // ---- mean(s*lam): wave32 shuffle sum + one barrier, fixed combine order ----
    float ws = wave_sum(sum);
    if (lane == 0) wred[wid] = ws;
    __syncthreads();
    float tot = 0.f;
    #pragma unroll
    for (int w = 0; w < NWAVES; ++w) tot += wred[w];
    const float mu_gap = tot * (1.0f / (float)QP_M);

    float rc_lo = 0.f, rc_hi = 0.f;
    if (tid < QP_M1) {
      rc_lo = s_lo * lam_lo - QP_SIGMA * mu_gap;
      rc_hi = s_hi * lam_hi - QP_SIGMA * mu_gap;
      float u_lo = (lam_lo * rp_lo - rc_lo) / s_lo;
      float u_hi = (lam_hi * rp_hi - rc_hi) / s_hi;
      shv2[tid] = u_lo - u_hi;                    // for G^T((lam*rp-rc)/s)
      shw[tid]  = lam_lo / s_lo + lam_hi / s_hi;  // wsum
    }
    __syncthreads();

    // ---- build tridiagonal M and rhs ----
    if (tid < QP_N) {
      float gtu = ((tid < QP_M1) ? shv2[tid] : 0.f) - ((tid > 0) ? shv2[tid - 1] : 0.f);
      float rhs = -(rd + gtu);
      float wl = (tid > 0)      ? shw[tid - 1] : 0.f;
      float wr = (tid < QP_M1)  ? shw[tid]     : 0.f;
      PA[0][tid] = -wl;
      PB[0][tid] = q + wl + wr;
      PC[0][tid] = -wr;
      PD[0][tid] = rhs;
    }
    __syncthreads();

    // ---- PCR: 8 steps (stride 1..128), strictly diagonally dominant ----
    int cur = 0;
    #pragma unroll
    for (int k = 1; k < QP_N; k <<= 1) {
      const int nxt = cur ^ 1;
      if (tid < QP_N) {
        float am, bm, cm, dm, ap, bp, cp, dp;
        pcr_read(PA[cur], PB[cur], PC[cur], PD[cur], tid - k, am, bm, cm, dm);
        pcr_read(PA[cur], PB[cur], PC[cur], PD[cur], tid + k, ap, bp, cp, dp);
        float a = PA[cur][tid], bb = PB[cur][tid], c = PC[cur][tid], dd = PD[cur][tid];
        float f1 = a / bm, f2 = c / bp;
        PA[nxt][tid] = -f1 * am;
        PC[nxt][tid] = -f2 * cp;
        PB[nxt][tid] = bb - f1 * cm - f2 * ap;
        PD[nxt][tid] = dd - f1 * dm - f2 * dp;
      }
      __syncthreads();
      cur = nxt;
    }

    float dz = 0.f;
    if (tid < QP_N) { dz = PD[cur][tid] / PB[cur][tid]; shdz[tid] = dz; }
    __syncthreads();

    // ---- ds, dlam, fraction-to-boundary ----
    float ds_lo = 0.f, ds_hi = 0.f, dlam_lo = 0.f, dlam_hi = 0.f;
    float amin = __builtin_inff();
    if (tid < QP_M1) {
      float ddz = shdz[tid] - shdz[tid + 1];
      ds_lo = -rp_lo - ddz;
      ds_hi = -rp_hi + ddz;
      dlam_lo = (-rc_lo - lam_lo * ds_lo) / s_lo;
      dlam_hi = (-rc_hi - lam_hi * ds_hi) / s_hi;
      if (ds_lo   < 0.f) amin = fminf(amin, -s_lo   / ds_lo);
      if (ds_hi   < 0.f) amin = fminf(amin, -s_hi   / ds_hi);
      if (dlam_lo < 0.f) amin = fminf(amin, -lam_lo / dlam_lo);
      if (dlam_hi < 0.f) amin = fminf(amin, -lam_hi / dlam_hi);
    }
    // ---- min over m: wave32 shuffle min + two barriers ----
    float wm = wave_min(amin);
    __syncthreads();                 // protect wred from previous use before rewrite
    if (lane == 0) wred[wid] = wm;
    __syncthreads();
    float gmin = __builtin_inff();
    #pragma unroll
    for (int w = 0; w < NWAVES; ++w) gmin = fminf(gmin, wred[w]);
    const float alpha = fminf(1.0f, 0.99f * gmin);

    if (tid < QP_N) z += alpha * dz;
    if (tid < QP_M1) {
      s_lo   += alpha * ds_lo;    s_hi   += alpha * ds_hi;
      lam_lo += alpha * dlam_lo;  lam_hi += alpha * dlam_hi;
    }
    __syncthreads();
  }

  if (tid < QP_N) out[base + tid] = z;
}

extern "C" void kernel_launch(void* const* d_in, const int* in_sizes, int n_in,
                              void* d_out, int out_size, void* d_ws, size_t ws_size,
                              hipStream_t stream) {
  (void)n_in; (void)out_size; (void)d_ws; (void)ws_size;
  const float* z0  = (const float*)d_in[0];
  const float* mu  = (const float*)d_in[1];
  const float* dg  = (const float*)d_in[2];
  const float* d2g = (const float*)d_in[3];
  float* out = (float*)d_out;
  const int nb = in_sizes[0] / QP_N;   // B = 256 batches -> one workgroup each
  qp_ipm_kernel<<<nb, BLK, 0, stream>>>(z0, mu, dg, d2g, out);
}